// NeuralPoint_63471026700449
// MI455X (gfx1250) — compile-verified
//
#include <hip/hip_runtime.h>
#include <stdint.h>

#define N_QUERY   32768
#define N_POINTS  8192
#define FEAT_DIM  32
#define KNN       8
#define CHUNK     2048     // points staged into LDS per pass (24KB raw xyz)
#define BLOCK     256      // 8 waves; 32 queries per wave -> 256 queries per block

typedef __attribute__((ext_vector_type(2))) float    v2f;
typedef __attribute__((ext_vector_type(8))) float    v8f;
typedef __attribute__((ext_vector_type(4))) unsigned u32x4;
typedef __attribute__((ext_vector_type(4))) int      i32x4;
typedef __attribute__((ext_vector_type(8))) int      i32x8;

#if defined(__has_builtin)
#if __has_builtin(__builtin_amdgcn_tensor_load_to_lds)
#define USE_TDM 1
#endif
#endif
#ifndef USE_TDM
#define USE_TDM 0
#endif

__global__ __launch_bounds__(BLOCK) void knn_wmma_kernel(
    const float* __restrict__ xyz, const float* __restrict__ points,
    const float* __restrict__ feats, float* __restrict__ out)
{
  __shared__ float sRaw[CHUNK * 3];    // raw staged points (x,y,z), 24KB
  __shared__ float sScore[8 * 512];    // per-wave 2x 16x16 score tiles, 16KB

  const int tid  = threadIdx.x;
  const int lane = tid & 31;
  const int wave = tid >> 5;
  const int h    = lane >> 4;   // which K-pair half of the wave
  const int n    = lane & 15;   // M (A) / N (B) position

  const int qwave = blockIdx.x * BLOCK + wave * 32;

  // ---- Build A fragments once: row m = [-2x, -2y, -2z, 1] ----
  // 16x4 f32 A: lanes 0-15 hold (K0,K1), lanes 16-31 hold (K2,K3) for M=n.
  const int qa = qwave + n;        // tile 0 queries
  const int qb = qwave + 16 + n;   // tile 1 queries
  float ax0 = xyz[3*qa], ay0 = xyz[3*qa+1], az0 = xyz[3*qa+2];
  float ax1 = xyz[3*qb], ay1 = xyz[3*qb+1], az1 = xyz[3*qb+2];
  v2f A0, A1;
  A0.x = h ? (-2.0f*az0) : (-2.0f*ax0);
  A0.y = h ? 1.0f        : (-2.0f*ay0);
  A1.x = h ? (-2.0f*az1) : (-2.0f*ax1);
  A1.y = h ? 1.0f        : (-2.0f*ay1);

  // ---- per-lane top-8 as packed sortable keys: (orderedU32(dist)<<32)|idx ----
  uint64_t k[KNN];
#pragma unroll
  for (int j = 0; j < KNN; ++j) k[j] = ~0ull;

  float*       sw  = sScore + wave * 512;          // tile0 @ [0,256), tile1 @ [256,512)
  const int    sb  = h * 128 + n;                  // C-spill base: element (m=8h, n)
  const float* row = sw + h * 256 + n * 16;        // this lane's query row (64B aligned)

  for (int base = 0; base < N_POINTS; base += CHUNK) {
    __syncthreads();                               // WAR vs previous pass readers

#if USE_TDM
    if (wave == 0) {
      // ---- Tensor Data Mover: 1-D copy of CHUNK*3 dwords into LDS ----
      uint64_t ga   = (uint64_t)(uintptr_t)points + (uint64_t)base * 12u;
      unsigned lds0 = (unsigned)(uint64_t)(__attribute__((address_space(3))) void*)sRaw;
      u32x4 g0;
      g0.x = 1u;                                        // count=1, user load
      g0.y = lds0;                                      // lds_addr [63:32]
      g0.z = (unsigned)(ga & 0xffffffffu);              // global_addr lo
      g0.w = (unsigned)((ga >> 32) & 0x01ffffffu)       // global_addr hi (57b)
           | (2u << 30);                                // type=2 ("image")
      i32x8 g1;
      g1[0] = (int)(2u << 16);                          // data_size=4B; mask/flags=0
      g1[1] = (int)(((CHUNK*3) & 0xffff) << 16);        // tensor_dim0[15:0] @ [63:48]
      g1[2] = (int)(((CHUNK*3) >> 16) | (1u << 16));    // tensor_dim0[31:16] | tensor_dim1=1
      g1[3] = (int)((unsigned)(CHUNK*3) << 16);         // tile_dim0 @ [127:112]
      g1[4] = 0; g1[5] = 0; g1[6] = 0; g1[7] = 0;       // tile_dim1=0 (1-D), strides=0
      i32x4 gz = {0, 0, 0, 0};
#if __clang_major__ >= 23
      i32x8 gz8 = {0, 0, 0, 0, 0, 0, 0, 0};
      __builtin_amdgcn_tensor_load_to_lds(g0, g1, gz, gz, gz8, 0);
#else
      __builtin_amdgcn_tensor_load_to_lds(g0, g1, gz, gz, 0);
#endif
      __builtin_amdgcn_s_wait_tensorcnt(0);
    }
    __syncthreads();
#else
    for (int t = tid; t < CHUNK * 3; t += BLOCK)
      sRaw[t] = points[(size_t)base * 3 + t];
    __syncthreads();
#endif

    for (int c = 0; c < CHUNK; c += 16) {
      // B fragment: 4x16 f32, K-rows striped over lane halves; col = [px,py,pz,||p||^2]
      const int pp = (c + n) * 3;
      float px = sRaw[pp], py = sRaw[pp + 1], pz = sRaw[pp + 2];
      v2f B;
      B.x = h ? pz : px;
      B.y = h ? (px*px + py*py + pz*pz) : py;

      v8f C0 = {}; v8f C1 = {};
      C0 = __builtin_amdgcn_wmma_f32_16x16x4_f32(false, A0, false, B,
                                                 (short)0, C0, false, false);
      C1 = __builtin_amdgcn_wmma_f32_16x16x4_f32(false, A1, false, B,
                                                 (short)0, C1, false, false);

      // Spill C tiles (lane L, vgpr v -> score[m=v+8h][n]) to transpose into rows.
#pragma unroll
      for (int v = 0; v < 8; ++v) {
        sw[sb + v*16]       = C0[v];
        sw[256 + sb + v*16] = C1[v];
      }
      // wave-private LDS RAW: wait all DS stores, block compiler reordering
      asm volatile("s_wait_dscnt 0" ::: "memory");

      // Pull this lane's 16 scores with 4x b128 loads, then update top-8.
      float r[16];
#pragma unroll
      for (int t4 = 0; t4 < 4; ++t4) {
        float4 v4 = ((const float4*)row)[t4];
        r[4*t4 + 0] = v4.x; r[4*t4 + 1] = v4.y;
        r[4*t4 + 2] = v4.z; r[4*t4 + 3] = v4.w;
      }
      const int pi0 = base + c;
#pragma unroll
      for (int t2 = 0; t2 < 16; ++t2) {
        unsigned su = __float_as_uint(r[t2]);
        su ^= ((unsigned)((int)su >> 31)) | 0x80000000u;   // monotone f32 -> u32
        uint64_t key = ((uint64_t)su << 32) | (unsigned)(pi0 + t2);
        if (key < k[KNN-1]) {                              // EXECZ-skippable filter
          bool cl[KNN];
#pragma unroll
          for (int j = 0; j < KNN; ++j) cl[j] = key < k[j];
#pragma unroll
          for (int j = KNN-1; j > 0; --j)
            k[j] = cl[j-1] ? k[j-1] : (cl[j] ? key : k[j]);
          if (cl[0]) k[0] = key;
        }
      }
    }
  }

  // ---- gather outputs: knn_points [N,8,3] then knn_feats [N,8,32] ----
  const int q = qwave + lane;
  float* outP = out + (size_t)q * (3 * KNN);
  float* outF = out + (size_t)N_QUERY * 3 * KNN + (size_t)q * (FEAT_DIM * KNN);
#pragma unroll
  for (int j = 0; j < KNN; ++j) {
    int p = (int)(unsigned)(k[j] & 0xffffffffu);
    outP[j*3 + 0] = points[3*p + 0];
    outP[j*3 + 1] = points[3*p + 1];
    outP[j*3 + 2] = points[3*p + 2];
    const float4* fp = (const float4*)(feats + (size_t)p * FEAT_DIM);
    float4*       op = (float4*)(outF + j * FEAT_DIM);
#pragma unroll
    for (int t = 0; t < FEAT_DIM/4; ++t) op[t] = fp[t];
  }
}

extern "C" void kernel_launch(void* const* d_in, const int* in_sizes, int n_in,
                              void* d_out, int out_size, void* d_ws, size_t ws_size,
                              hipStream_t stream) {
  (void)in_sizes; (void)n_in; (void)d_ws; (void)ws_size; (void)out_size;
  const float* xyz    = (const float*)d_in[0];
  const float* points = (const float*)d_in[1];
  const float* feats  = (const float*)d_in[2];
  float* out = (float*)d_out;

  dim3 grid(N_QUERY / BLOCK);   // 128 blocks x 256 threads (8 waves each)
  knn_wmma_kernel<<<grid, BLOCK, 0, stream>>>(xyz, points, feats, out);
}